// DenstiyRatioEstimator_52948356825617
// MI455X (gfx1250) — compile-verified
//
#include <hip/hip_runtime.h>
#include <math.h>

// Problem sizes (fixed by the reference).
#define B_  2048
#define L_  128
#define Z_  256
#define C_  256
#define CHUNKS 16
#define BPC (B_ / CHUNKS)   // 128 batch rows per chunk

typedef __attribute__((ext_vector_type(2))) float v2f;
typedef __attribute__((ext_vector_type(8))) float v8f;

// ---------------------------------------------------------------------------
// K1: partial csum over b-chunks.  grid = (L_, CHUNKS), block = 256 (== C_).
// Reads all of c (256 MB) -> the dominant-bandwidth kernel alongside K4.
// 16384 waves total; 4 independent accumulators so loads pipeline instead
// of serializing on one FADD chain.
// ---------------------------------------------------------------------------
__global__ void k_csum_partial(const float* __restrict__ c,
                               float* __restrict__ partial) {
    const int l   = blockIdx.x;    // 0..127
    const int cb  = blockIdx.y;    // 0..15
    const int tid = threadIdx.x;   // 0..255  == channel index
    const size_t rowstride = (size_t)L_ * C_;
    const float* p = c + ((size_t)(cb * BPC) * L_ + l) * C_ + tid;

    float a0 = 0.f, a1 = 0.f, a2 = 0.f, a3 = 0.f;
    #pragma unroll 4
    for (int b = 0; b < BPC; b += 4) {
        a0 += p[(size_t)(b + 0) * rowstride];
        a1 += p[(size_t)(b + 1) * rowstride];
        a2 += p[(size_t)(b + 2) * rowstride];
        a3 += p[(size_t)(b + 3) * rowstride];
    }
    partial[((size_t)cb * L_ + l) * C_ + tid] = (a0 + a1) + (a2 + a3);
}

// ---------------------------------------------------------------------------
// K2: fold the 16 partials -> csum (L_ x C_).  grid = L_, block = 256.
// ---------------------------------------------------------------------------
__global__ void k_csum_final(const float* __restrict__ partial,
                             float* __restrict__ csum) {
    const int l = blockIdx.x, tid = threadIdx.x;
    float acc = 0.f;
    #pragma unroll
    for (int cb = 0; cb < CHUNKS; ++cb)
        acc += partial[((size_t)cb * L_ + l) * C_ + tid];
    csum[(size_t)l * C_ + tid] = acc;
}

// ---------------------------------------------------------------------------
// K3: v[l, z] = sum_c trans[l,z,c] * csum[l,c]   via V_WMMA_F32_16X16X4_F32.
// grid = (Z_/16, L_), block = 32 (one full wave -> EXEC all ones for WMMA).
// GEMV as GEMM: B's 16 columns all hold the same csum chunk, so every column
// of D holds the 16-long result chunk.
//   A (16x4 f32) layout (ISA 7.12.2): v0 = K0 (lanes 0-15) / K2 (lanes 16-31),
//                                     v1 = K1 / K3.
//   B (4x16 f32): rows striped across lanes, upper half-wave = row +2
//   (same +rows/2 pattern as the documented C/D layout); broadcast makes the
//   N mapping irrelevant.
// ---------------------------------------------------------------------------
__global__ void k_v_wmma(const float* __restrict__ trans,
                         const float* __restrict__ csum,
                         float* __restrict__ v) {
    const int zt   = blockIdx.x;          // z tile, 0..15
    const int l    = blockIdx.y;          // 0..127
    const int lane = threadIdx.x;         // 0..31
    const int half = lane >> 4;           // 0: K0/K1, 1: K2/K3
    const int row  = lane & 15;           // M index within tile

    const float* tbase = trans + ((size_t)l * Z_ + zt * 16 + row) * C_ + half * 2;
    const float* cbase = csum + (size_t)l * C_ + half * 2;

    v8f acc = {};
    for (int c0 = 0; c0 < C_; c0 += 4) {
        v2f a, b;
        a.x = tbase[c0];     // A[M=row][K = c0 + 2*half]
        a.y = tbase[c0 + 1]; // A[M=row][K = c0 + 2*half + 1]
        b.x = cbase[c0];     // B row (c0 + 2*half), broadcast across N
        b.y = cbase[c0 + 1]; // B row (c0 + 2*half + 1)
        acc = __builtin_amdgcn_wmma_f32_16x16x4_f32(
            /*neg_a=*/false, a, /*neg_b=*/false, b,
            /*c_mod=*/(short)0, acc, /*reuse_a=*/false, /*reuse_b=*/false);
    }
    // D layout: VGPR p -> M=p (lanes 0-15), M=p+8 (lanes 16-31); all N equal.
    if (row == 0) {  // lanes 0 and 16 write M=0..7 and M=8..15
        float* out = v + (size_t)l * Z_ + zt * 16 + half * 8;
        #pragma unroll
        for (int p = 0; p < 8; ++p) out[p] = acc[p];
    }
}

// ---------------------------------------------------------------------------
// K4: out[b,l] = exp( dot(z[b,l,:], v[l,:]) ).  One wave per output row.
// grid = B_*L_/8, block = 256 (8 waves).  Streams z (256 MB) with float4
// coalesced loads; v is 128 KB and stays cache-resident.
// ---------------------------------------------------------------------------
__global__ void k_dot_exp(const float* __restrict__ z,
                          const float* __restrict__ v,
                          float* __restrict__ out) {
    const int lane = threadIdx.x & 31;
    const int wave = threadIdx.x >> 5;
    const size_t row = (size_t)blockIdx.x * 8 + wave;   // row in (B*L)
    const int l = (int)(row & (L_ - 1));

    const float4* zp = (const float4*)(z + row * Z_);
    const float4* vp = (const float4*)(v + (size_t)l * Z_);

    float acc = 0.f;
    #pragma unroll
    for (int i = 0; i < Z_ / 4 / 32; ++i) {             // 2 iterations
        const float4 zz = zp[lane + i * 32];
        const float4 vv = vp[lane + i * 32];
        acc += zz.x * vv.x + zz.y * vv.y + zz.z * vv.z + zz.w * vv.w;
    }
    #pragma unroll
    for (int off = 16; off > 0; off >>= 1)
        acc += __shfl_xor(acc, off, 32);
    if (lane == 0) out[row] = expf(acc);
}

// ---------------------------------------------------------------------------
extern "C" void kernel_launch(void* const* d_in, const int* in_sizes, int n_in,
                              void* d_out, int out_size, void* d_ws, size_t ws_size,
                              hipStream_t stream) {
    const float* z     = (const float*)d_in[0];  // (B, L, Z)
    const float* c     = (const float*)d_in[1];  // (B, L, C)
    const float* trans = (const float*)d_in[2];  // (L, Z, C)
    float* out = (float*)d_out;                  // (B, L)

    // Workspace layout (floats): [partial | csum | v]
    float* partial = (float*)d_ws;                         // CHUNKS*L_*C_ = 524288
    float* csum    = partial + (size_t)CHUNKS * L_ * C_;   // L_*C_        =  32768
    float* v       = csum + (size_t)L_ * C_;               // L_*Z_        =  32768

    k_csum_partial<<<dim3(L_, CHUNKS), 256, 0, stream>>>(c, partial);
    k_csum_final<<<L_, 256, 0, stream>>>(partial, csum);
    k_v_wmma<<<dim3(Z_ / 16, L_), 32, 0, stream>>>(trans, csum, v);
    k_dot_exp<<<(B_ * L_) / 8, 256, 0, stream>>>(z, v, out);
}